// RepulsionEnergy_18562848654089
// MI455X (gfx1250) — compile-verified
//
#include <hip/hip_runtime.h>
#include <cstdint>

#ifndef __has_builtin
#define __has_builtin(x) 0
#endif

#if __has_builtin(__builtin_amdgcn_global_load_async_to_lds_b128) && \
    __has_builtin(__builtin_amdgcn_s_wait_asynccnt)
#define USE_ASYNC_LDS 1
#else
#define USE_ASYNC_LDS 0
#endif

#define THREADS 256
#define EPT 4                      // edges per thread per tile (one int4)
#define TILE (THREADS * EPT)       // 1024 edges per tile
#define MAX_BLOCKS 2048

// cpol: TH in [2:0]; 1 = TH_LOAD_NT (non-temporal) -- edge stream is read-once,
// keep it from evicting the L2-resident batch/pos/z gather tables.
#define CPOL_NT 1

#define AA2AU_F 1.8897261258369282f
#define AU2KCAL_F 627.5094740630558f
#define CUT2_F 9.0f                // CUTOFF^2

#if USE_ASYNC_LDS
// Builtin signature (from clang diagnostic): param0 = v4i in AS1 (prints as
// "int4 __device__ *"), param1 = v4i in AS3, then imm offset + imm cpol.
typedef int v4i __attribute__((vector_size(16)));
typedef __attribute__((address_space(1))) v4i gv4i;
typedef __attribute__((address_space(3))) v4i lv4i;

__device__ __forceinline__ gv4i* as_global_v4(const void* p) {
    return (gv4i*)(unsigned long long)(uintptr_t)p;
}
__device__ __forceinline__ lv4i* as_lds_v4(void* p) {
    // flat LDS address: low 32 bits are the LDS byte offset on gfx1250
    return (lv4i*)(unsigned int)(uintptr_t)p;
}
#endif

__device__ __forceinline__ void edge_contrib(
    int s, int t,
    const float* __restrict__ pos, const int* __restrict__ z,
    const float* __restrict__ arep, const float* __restrict__ zeff,
    const int* __restrict__ batch, float* __restrict__ out)
{
    // Hot path: only two 4B gathers from the L2-resident batch table.
    int bs = batch[s];
    int bt = batch[t];
    if (bs != bt) return;                      // ~99.95% of edges exit here
    float dx = pos[3 * s + 0] - pos[3 * t + 0];
    float dy = pos[3 * s + 1] - pos[3 * t + 1];
    float dz = pos[3 * s + 2] - pos[3 * t + 2];
    float d2 = dx * dx + dy * dy + dz * dz;
    if (d2 > CUT2_F) return;                   // d <= CUTOFF (d>=1e-9 clamp below)
    float d   = fmaxf(sqrtf(d2), 1e-9f);
    float dau = d * AA2AU_F;
    int zs = z[s], zt = z[t];
    float a  = sqrtf(arep[zs] * arep[zt]);
    float ex = __expf(-a * dau * sqrtf(dau));  // dau^1.5 = dau*sqrt(dau)
    float rep = zeff[zs] * zeff[zt] * ex / dau * AU2KCAL_F;
    atomicAdd(&out[bs], rep);                  // ~6K total hits -> negligible
}

__global__ void zero_kernel(float* __restrict__ out, int n) {
    int i = blockIdx.x * blockDim.x + threadIdx.x;
    if (i < n) out[i] = 0.0f;
}

__global__ void __launch_bounds__(THREADS)
repulsion_kernel(const float* __restrict__ pos,
                 const float* __restrict__ arep,
                 const float* __restrict__ zeff,
                 const int*  __restrict__ z,
                 const int*  __restrict__ esrc,
                 const int*  __restrict__ edst,
                 const int*  __restrict__ batch,
                 float* __restrict__ out,
                 long long E)
{
    __shared__ int4 s_src[2][THREADS];
    __shared__ int4 s_dst[2][THREADS];

    const int tid = threadIdx.x;
    const long long numTiles = E / TILE;

    long long tile = blockIdx.x;
    int cur = 0;
    bool have = (tile < numTiles);

    if (have) {
        const int* gs = esrc + tile * TILE + tid * EPT;
        const int* gd = edst + tile * TILE + tid * EPT;
#if USE_ASYNC_LDS
        __builtin_amdgcn_global_load_async_to_lds_b128(as_global_v4(gs), as_lds_v4(&s_src[0][tid]), 0, CPOL_NT);
        __builtin_amdgcn_global_load_async_to_lds_b128(as_global_v4(gd), as_lds_v4(&s_dst[0][tid]), 0, CPOL_NT);
#else
        s_src[0][tid] = *(const int4*)gs;
        s_dst[0][tid] = *(const int4*)gd;
#endif
    }

    while (have) {
        long long nxt = tile + gridDim.x;
        bool haveNext = (nxt < numTiles);
        if (haveNext) {
            const int* gs = esrc + nxt * TILE + tid * EPT;
            const int* gd = edst + nxt * TILE + tid * EPT;
            // deep prefetch of the tile after next (stay in-bounds, streaming)
            long long pf = nxt + gridDim.x;
            if (pf < numTiles) {
                __builtin_prefetch(esrc + pf * TILE + tid * EPT, 0, 0);
                __builtin_prefetch(edst + pf * TILE + tid * EPT, 0, 0);
            }
#if USE_ASYNC_LDS
            __builtin_amdgcn_global_load_async_to_lds_b128(as_global_v4(gs), as_lds_v4(&s_src[cur ^ 1][tid]), 0, CPOL_NT);
            __builtin_amdgcn_global_load_async_to_lds_b128(as_global_v4(gd), as_lds_v4(&s_dst[cur ^ 1][tid]), 0, CPOL_NT);
#else
            s_src[cur ^ 1][tid] = *(const int4*)gs;
            s_dst[cur ^ 1][tid] = *(const int4*)gd;
#endif
        }
#if USE_ASYNC_LDS
        // async loads complete in order per wave: <=2 outstanding means the
        // 2 loads for the current buffer have landed in LDS.
        if (haveNext) {
            __builtin_amdgcn_s_wait_asynccnt(2);
        } else {
            __builtin_amdgcn_s_wait_asynccnt(0);
        }
#endif
        int4 sv = s_src[cur][tid];
        int4 dv = s_dst[cur][tid];
        edge_contrib(sv.x, dv.x, pos, z, arep, zeff, batch, out);
        edge_contrib(sv.y, dv.y, pos, z, arep, zeff, batch, out);
        edge_contrib(sv.z, dv.z, pos, z, arep, zeff, batch, out);
        edge_contrib(sv.w, dv.w, pos, z, arep, zeff, batch, out);

        cur ^= 1;
        tile = nxt;
        have = haveNext;
    }

    // tail edges (E not a multiple of TILE)
    long long tailStart = numTiles * TILE;
    for (long long i = tailStart + (long long)blockIdx.x * THREADS + tid;
         i < E;
         i += (long long)gridDim.x * THREADS) {
        edge_contrib(esrc[i], edst[i], pos, z, arep, zeff, batch, out);
    }
}

extern "C" void kernel_launch(void* const* d_in, const int* in_sizes, int n_in,
                              void* d_out, int out_size, void* d_ws, size_t ws_size,
                              hipStream_t stream) {
    const float* pos   = (const float*)d_in[0];
    const float* arep  = (const float*)d_in[1];
    const float* zeff  = (const float*)d_in[2];
    const int*   z     = (const int*)d_in[3];
    const int*   eidx  = (const int*)d_in[4];
    const int*   batch = (const int*)d_in[5];
    float* out = (float*)d_out;

    long long E = (long long)in_sizes[4] / 2;   // edge_index is [2, E]
    const int* esrc = eidx;
    const int* edst = eidx + E;

    zero_kernel<<<(out_size + 255) / 256, 256, 0, stream>>>(out, out_size);

    long long numTiles = E / TILE;
    long long want = numTiles > 0 ? numTiles : 1;
    int blocks = (int)(want < MAX_BLOCKS ? want : MAX_BLOCKS);
    repulsion_kernel<<<blocks, THREADS, 0, stream>>>(
        pos, arep, zeff, z, esrc, edst, batch, out, E);
}